// MeanMemoryMessageReducer_62775241999071
// MI455X (gfx1250) — compile-verified
//
#include <hip/hip_runtime.h>

typedef __attribute__((ext_vector_type(2))) float v2f;
typedef __attribute__((ext_vector_type(8))) float v8f;

// lower_bound over sorted int32 segment ids (8 MB array -> L2-resident)
__device__ __forceinline__ int lower_bound_i32(const int* __restrict__ a, int n, int v) {
    int lo = 0, hi = n;
    while (lo < hi) {
        int m = (lo + hi) >> 1;
        if (a[m] < v) lo = m + 1; else hi = m;
    }
    return lo;
}

// One block per segment; wave w owns columns [16w,16w+16).
// Segment column-sum via V_WMMA_F32_16X16X4_F32 with A = ones(16x4):
// B (4x16 f32, 2 VGPRs): VGPR0 = rows {r, r+2}, VGPR1 = rows {r+1, r+3},
// lanes 0-15 handle rows {r,r+1}, lanes 16-31 rows {r+2,r+3}.
// DD > 0: compile-time D (row stride folds into immediate load offsets).
template<int DD>
__global__ __launch_bounds__(256) void seg_mean_wmma_kernel(
    const float* __restrict__ messages,
    const float* __restrict__ timestamps,
    const int*   __restrict__ seg_ids,
    float* __restrict__ out,          // layout: [M] nids | [M*D] means | [M] ts
    int N, int M, int Dparam)
{
    const int D = (DD > 0) ? DD : Dparam;
    const int s = blockIdx.x;

    __shared__ int s_bounds[2];
    if (threadIdx.x < 2)
        s_bounds[threadIdx.x] = lower_bound_i32(seg_ids, N, s + (int)threadIdx.x);
    __syncthreads();
    const int start = s_bounds[0];
    const int end   = s_bounds[1];
    const int count = end - start;
    const int nfull = count & ~3;      // rows handled by the branch-free WMMA loop

    const int lane = threadIdx.x & 31;
    const int wave = threadIdx.x >> 5;
    const int half = lane >> 4;        // 0: rows {r,r+1}; 1: rows {r+2,r+3}
    const int col  = (wave << 4) + (lane & 15);

    v2f a; a[0] = 1.0f; a[1] = 1.0f;   // ones A-matrix (16x4)
    v8f c = {};                        // f32 C/D accumulator

    // lane-private stream pointer; with DD=128 the second row is a constant
    // 512-byte immediate offset (p[D]) -> both loads share one vaddr, and the
    // 2048-byte tile stride is an inline-constant 64-bit add.
    const float* p = messages + (size_t)(start + (half << 1)) * D + col;
    const size_t step = (size_t)4 * D;

    if (nfull > 0) {
        v2f b; b[0] = p[0]; b[1] = p[D]; p += step;          // preload tile 0
        for (int r = 4; r < nfull; r += 4) {
            v2f bn; bn[0] = p[0]; bn[1] = p[D]; p += step;   // load tile i+1
            c = __builtin_amdgcn_wmma_f32_16x16x4_f32(       // WMMA on tile i
                    false, a, false, b, (short)0, c, false, false);
            b = bn;
        }
        c = __builtin_amdgcn_wmma_f32_16x16x4_f32(
                false, a, false, b, (short)0, c, false, false);
    }

    // Tail (count % 4 rows): block-uniform trip count, branch-free scalar adds.
    float tacc = 0.0f;
    for (int r = start + nfull; r < end; ++r)
        tacc += messages[(size_t)r * D + col];

    if (lane < 16) {
        // D-matrix VGPR0, lanes 0-15 = row M=0, N=lane -> full column sum
        const float inv = 1.0f / (float)(count > 0 ? count : 1);
        out[(size_t)M + (size_t)s * D + col] = (c[0] + tacc) * inv;
    }
    if (threadIdx.x == 0) {
        out[s] = (float)s;                              // unique_nids = arange(M)
        const int li = (count > 0) ? (end - 1) : 0;     // segment_max(pos), clipped
        out[(size_t)M + (size_t)M * D + s] = timestamps[li];
    }
}

extern "C" void kernel_launch(void* const* d_in, const int* in_sizes, int n_in,
                              void* d_out, int out_size, void* d_ws, size_t ws_size,
                              hipStream_t stream) {
    // setup_inputs order: unique_nids, messages, timestamps, segment_ids, num_segments
    const float* messages   = (const float*)d_in[1];
    const float* timestamps = (const float*)d_in[2];
    const int*   seg_ids    = (const int*)d_in[3];

    const int M = in_sizes[0];
    const int N = in_sizes[2];
    const int D = in_sizes[1] / N;          // 128 for this problem

    const int block = (D / 16) * 32;        // one wave per 16-column slice -> 256
    if (D == 128) {
        seg_mean_wmma_kernel<128><<<M, block, 0, stream>>>(
            messages, timestamps, seg_ids, (float*)d_out, N, M, D);
    } else {
        seg_mean_wmma_kernel<0><<<M, block, 0, stream>>>(
            messages, timestamps, seg_ids, (float*)d_out, N, M, D);
    }
}